// LightGCN_51419348468279
// MI455X (gfx1250) — compile-verified
//
#include <hip/hip_runtime.h>
#include <hip/hip_bf16.h>

#define DIM 64

typedef float v2f __attribute__((ext_vector_type(2)));
typedef float v8f __attribute__((ext_vector_type(8)));

// ---------------------------------------------------------------------------
// init: all_emb = concat(user_emb, item_emb); cur = acc = all_emb
// ---------------------------------------------------------------------------
__global__ void lgcn_init(const float* __restrict__ uw, const float* __restrict__ iw,
                          float* __restrict__ cur, float* __restrict__ acc,
                          int U, int N) {
    int t = blockIdx.x * blockDim.x + threadIdx.x;   // one float4 per thread
    int total = N * (DIM / 4);
    if (t >= total) return;
    int row = t / (DIM / 4);
    int c4  = t % (DIM / 4);
    const float4* src = (row < U)
        ? (const float4*)(uw + (size_t)row * DIM)
        : (const float4*)(iw + (size_t)(row - U) * DIM);
    float4 v = src[c4];
    ((float4*)(cur + (size_t)row * DIM))[c4] = v;
    ((float4*)(acc + (size_t)row * DIM))[c4] = v;
}

// ---------------------------------------------------------------------------
// zero a float buffer (graph-capture-safe replacement for hipMemsetAsync)
// ---------------------------------------------------------------------------
__global__ void lgcn_zero(float4* __restrict__ p, int total4) {
    int t = blockIdx.x * blockDim.x + threadIdx.x;
    if (t < total4) p[t] = make_float4(0.f, 0.f, 0.f, 0.f);
}

// ---------------------------------------------------------------------------
// SpMM: y[dst,:] += val * x[src,:]. 16 lanes per edge, float4 per lane.
// x (38.4 MB) is L2-resident; atomics land in L2.
// ---------------------------------------------------------------------------
__global__ void lgcn_spmm(const int* __restrict__ esrc, const int* __restrict__ edst,
                          const float* __restrict__ eval,
                          const float* __restrict__ x, float* __restrict__ y,
                          int E) {
    long long t = (long long)blockIdx.x * blockDim.x + threadIdx.x;
    long long e = t >> 4;
    if (e >= E) return;
    int lane16 = (int)(t & 15);
    int s = esrc[e];
    int d = edst[e];
    float v = eval[e];
    float4 xv = ((const float4*)(x + (size_t)s * DIM))[lane16];
    float* yp = y + (size_t)d * DIM + lane16 * 4;
    unsafeAtomicAdd(yp + 0, v * xv.x);
    unsafeAtomicAdd(yp + 1, v * xv.y);
    unsafeAtomicAdd(yp + 2, v * xv.z);
    unsafeAtomicAdd(yp + 3, v * xv.w);
}

// ---------------------------------------------------------------------------
// acc += layer
// ---------------------------------------------------------------------------
__global__ void lgcn_accum(float4* __restrict__ acc, const float4* __restrict__ add,
                           int total4) {
    int t = blockIdx.x * blockDim.x + threadIdx.x;
    if (t >= total4) return;
    float4 a = acc[t], b = add[t];
    a.x += b.x; a.y += b.y; a.z += b.z; a.w += b.w;
    acc[t] = a;
}

// ---------------------------------------------------------------------------
// Batched dot scores via V_WMMA_F32_16X16X4_F32.
// One wave (32 lanes) per 16 batch rows; accumulate 16x16 tile of U*P^T over
// K=64 in 16 steps of K=4; extract diagonal; scale by 1/(L+1)^2 = 1/16.
// A layout (16x4 f32): lanes 0-15 -> K=k0,k0+1 ; lanes 16-31 -> K=k0+2,k0+3,
// row M = lane&15. B (4x16) mirrored per column N = lane&15.
// EXEC all-1s around the WMMA (no divergence in the loop).
// ---------------------------------------------------------------------------
__global__ void __launch_bounds__(32)
lgcn_score(const float* __restrict__ acc,
           const int* __restrict__ bu, const int* __restrict__ bp,
           const int* __restrict__ bn,
           float* __restrict__ out_pos, float* __restrict__ out_neg, int U) {
    __shared__ float lp[32][8];
    __shared__ float ln[32][8];

    int g    = blockIdx.x;
    int lane = threadIdx.x;
    int r    = lane & 15;
    int hi   = lane >> 4;
    int gm   = g * 16 + r;

    const float* urow = acc + (size_t)bu[gm] * DIM;
    const float* prow = acc + (size_t)(U + bp[gm]) * DIM;
    const float* nrow = acc + (size_t)(U + bn[gm]) * DIM;

    v8f cp = {};
    v8f cn = {};
#pragma unroll
    for (int k0 = 0; k0 < DIM; k0 += 4) {
        int k = k0 + 2 * hi;                    // even -> 8B aligned
        v2f a  = *(const v2f*)(urow + k);
        v2f bpv = *(const v2f*)(prow + k);
        v2f bnv = *(const v2f*)(nrow + k);
        cp = __builtin_amdgcn_wmma_f32_16x16x4_f32(false, a, false, bpv,
                                                   (short)0, cp, false, false);
        cn = __builtin_amdgcn_wmma_f32_16x16x4_f32(false, a, false, bnv,
                                                   (short)0, cn, false, false);
    }

#pragma unroll
    for (int i = 0; i < 8; ++i) { lp[lane][i] = cp[i]; ln[lane][i] = cn[i]; }
    __syncthreads();

    if (lane < 16) {
        int m = lane;
        float ps, ns;
        if (m < 8) { ps = lp[m][m];          ns = ln[m][m]; }
        else       { ps = lp[16 + m][m - 8]; ns = ln[16 + m][m - 8]; }
        out_pos[gm] = ps * 0.0625f;   // (acc/4)·(acc/4) = acc·acc / 16
        out_neg[gm] = ns * 0.0625f;
    }
}

// ---------------------------------------------------------------------------
// reg_loss = sum |u0|^2 + |p0|^2 + |n0|^2 over the batch (ego embeddings)
// ---------------------------------------------------------------------------
__global__ void lgcn_zero_scalar(float* p) {
    if (threadIdx.x == 0 && blockIdx.x == 0) *p = 0.f;
}

__global__ void lgcn_reg(const float* __restrict__ uw, const float* __restrict__ iw,
                         const int* __restrict__ bu, const int* __restrict__ bp,
                         const int* __restrict__ bn, float* __restrict__ out, int B) {
    int t = blockIdx.x * blockDim.x + threadIdx.x;
    float s = 0.f;
    if (t < B) {
        const float4* u = (const float4*)(uw + (size_t)bu[t] * DIM);
        const float4* p = (const float4*)(iw + (size_t)bp[t] * DIM);
        const float4* n = (const float4*)(iw + (size_t)bn[t] * DIM);
#pragma unroll 4
        for (int i = 0; i < DIM / 4; ++i) {
            float4 a = u[i], b = p[i], c = n[i];
            s += a.x*a.x + a.y*a.y + a.z*a.z + a.w*a.w;
            s += b.x*b.x + b.y*b.y + b.z*b.z + b.w*b.w;
            s += c.x*c.x + c.y*c.y + c.z*c.z + c.w*c.w;
        }
    }
    __shared__ float red[256];
    red[threadIdx.x] = s;
    __syncthreads();
    for (int off = 128; off > 0; off >>= 1) {
        if (threadIdx.x < off) red[threadIdx.x] += red[threadIdx.x + off];
        __syncthreads();
    }
    if (threadIdx.x == 0) unsafeAtomicAdd(out, red[0]);
}

// ---------------------------------------------------------------------------
extern "C" void kernel_launch(void* const* d_in, const int* in_sizes, int n_in,
                              void* d_out, int out_size, void* d_ws, size_t ws_size,
                              hipStream_t stream) {
    const int*   esrc = (const int*)  d_in[0];
    const int*   edst = (const int*)  d_in[1];
    const float* eval = (const float*)d_in[2];
    const float* uw   = (const float*)d_in[3];
    const float* iw   = (const float*)d_in[4];
    const int*   bu   = (const int*)  d_in[5];
    const int*   bp   = (const int*)  d_in[6];
    const int*   bn   = (const int*)  d_in[7];

    const int E = in_sizes[0];
    const int U = in_sizes[3] / DIM;
    const int I = in_sizes[4] / DIM;
    const int N = U + I;
    const int B = in_sizes[5];

    float* out = (float*)d_out;
    size_t nd  = (size_t)N * DIM;
    float* cur = (float*)d_ws;       // layer input
    float* nxt = cur + nd;           // layer output
    float* acc = nxt + nd;           // running sum of layer embeddings

    const int total4 = N * (DIM / 4);
    const int zb  = (total4 + 255) / 256;
    const long long spmm_threads = (long long)E * 16;
    const int spmm_blocks = (int)((spmm_threads + 255) / 256);

    lgcn_init<<<zb, 256, 0, stream>>>(uw, iw, cur, acc, U, N);

    float* a = cur;
    float* b = nxt;
    for (int l = 0; l < 3; ++l) {
        lgcn_zero<<<zb, 256, 0, stream>>>((float4*)b, total4);
        lgcn_spmm<<<spmm_blocks, 256, 0, stream>>>(esrc, edst, eval, a, b, E);
        lgcn_accum<<<zb, 256, 0, stream>>>((float4*)acc, (const float4*)b, total4);
        float* t = a; a = b; b = t;
    }

    lgcn_score<<<B / 16, 32, 0, stream>>>(acc, bu, bp, bn, out, out + B, U);

    lgcn_zero_scalar<<<1, 32, 0, stream>>>(out + 2 * (size_t)B);
    lgcn_reg<<<(B + 255) / 256, 256, 0, stream>>>(uw, iw, bu, bp, bn,
                                                  out + 2 * (size_t)B, B);
}